// CustomGATHeadLayer_25632364822805
// MI455X (gfx1250) — compile-verified
//
#include <hip/hip_runtime.h>
#include <hip/hip_bf16.h>

#define N_NODES 100000
#define N_EDGES 1600000
#define IN_DIM 128
#define OUT_DIM 32
#define BN_EPS 1e-5f
#define LRELU_SLOPE 0.01f

typedef __attribute__((ext_vector_type(2))) float v2f;
typedef __attribute__((ext_vector_type(8))) float v8f;

// ---- ordered-uint encoding for exact float atomic max ----
__device__ __forceinline__ unsigned fflip(float f) {
    unsigned u = __float_as_uint(f);
    return (u >> 31) ? ~u : (u | 0x80000000u);
}
__device__ __forceinline__ float funflip(unsigned u) {
    u = (u >> 31) ? (u & 0x7FFFFFFFu) : ~u;
    return __uint_as_float(u);
}

// ---- 0: init accumulators ----
__global__ void gat_init_kernel(float* hout, float* esum, unsigned* emax, float* bn) {
    int idx = blockIdx.x * blockDim.x + threadIdx.x;
    int stride = gridDim.x * blockDim.x;
    for (int i = idx; i < N_NODES * OUT_DIM; i += stride) hout[i] = 0.0f;
    for (int i = idx; i < N_NODES; i += stride) { esum[i] = 0.0f; emax[i] = 0u; } // 0 encodes < -inf
    for (int i = idx; i < 2 * OUT_DIM; i += stride) bn[i] = 0.0f;
}

// ---- 1: z = h @ W_fc via V_WMMA_F32_16X16X4_F32 ----
// One wave handles a 16-row tile; N=32 -> two 16x16 accumulators; K=128 -> 32 wmma steps.
// A 16x4 f32 layout: lanes 0-15 = rows, vgpr0/1 = K0/K1; lanes 16-31 = rows, K2/K3.
// B 4x16 f32 layout mirrored: lanes 0-15 = cols with K0/K1; lanes 16-31 = cols with K2/K3.
// C/D: vgpr r, lanes 0-15 -> M=r, N=lane; lanes 16-31 -> M=r+8, N=lane-16.
__global__ void gat_gemm_wmma_kernel(const float* __restrict__ h,
                                     const float* __restrict__ W,
                                     float* __restrict__ z) {
    __shared__ float lds_w[IN_DIM * OUT_DIM]; // 16 KB, W[k*32+n]
    for (int i = threadIdx.x; i < IN_DIM * OUT_DIM; i += blockDim.x)
        lds_w[i] = W[i];
    __syncthreads();

    const int wave = threadIdx.x >> 5;
    const int lane = threadIdx.x & 31;
    const int tile = blockIdx.x * (blockDim.x >> 5) + wave;   // wave-uniform
    if (tile >= N_NODES / 16) return;                          // EXEC stays all-ones

    const int m0   = tile * 16;
    const int row  = m0 + (lane & 15);
    const int koff = (lane >> 4) * 2;   // 0 for lanes 0-15, 2 for lanes 16-31
    const int col  = lane & 15;

    v8f c0 = {}, c1 = {};
    const float* hrow = h + (size_t)row * IN_DIM;

    #pragma unroll 4
    for (int k = 0; k < IN_DIM; k += 4) {
        // A fragment: 2 consecutive f32 from this lane's row
        v2f a;
        a.x = hrow[k + koff];
        a.y = hrow[k + koff + 1];
        // B fragments from LDS (W is row-major k x n)
        v2f b0, b1;
        b0.x = lds_w[(k + koff)     * OUT_DIM + col];
        b0.y = lds_w[(k + koff + 1) * OUT_DIM + col];
        b1.x = lds_w[(k + koff)     * OUT_DIM + col + 16];
        b1.y = lds_w[(k + koff + 1) * OUT_DIM + col + 16];
        c0 = __builtin_amdgcn_wmma_f32_16x16x4_f32(false, a, false, b0, (short)0, c0, false, false);
        c1 = __builtin_amdgcn_wmma_f32_16x16x4_f32(false, a, false, b1, (short)0, c1, false, false);
    }

    const int rbase = m0 + ((lane >> 4) * 8);
    #pragma unroll
    for (int r = 0; r < 8; ++r) {
        z[(size_t)(rbase + r) * OUT_DIM + col]      = c0[r];
        z[(size_t)(rbase + r) * OUT_DIM + col + 16] = c1[r];
    }
}

// ---- 2: el = z . a_l, er = z . a_r (lane-per-channel, wave32 reduce) ----
__global__ void gat_attn_proj_kernel(const float* __restrict__ z,
                                     const float* __restrict__ Wa,
                                     float* __restrict__ el,
                                     float* __restrict__ er) {
    int gid = blockIdx.x * blockDim.x + threadIdx.x;
    int node = gid >> 5;
    int lane = gid & 31;
    if (node >= N_NODES) return;
    float v  = z[(size_t)node * OUT_DIM + lane];
    float pl = v * Wa[lane];
    float pr = v * Wa[OUT_DIM + lane];
    #pragma unroll
    for (int off = 16; off > 0; off >>= 1) {
        pl += __shfl_down(pl, off, 32);
        pr += __shfl_down(pr, off, 32);
    }
    if (lane == 0) { el[node] = pl; er[node] = pr; }
}

// ---- 3: e = lrelu(el[src]+er[dst]); segment max over dst ----
__global__ void gat_edge_max_kernel(const float* __restrict__ el,
                                    const float* __restrict__ er,
                                    const int* __restrict__ src,
                                    const int* __restrict__ dst,
                                    float* __restrict__ e_edge,
                                    unsigned* __restrict__ emax) {
    int i = blockIdx.x * blockDim.x + threadIdx.x;
    if (i >= N_EDGES) return;
    float e = el[src[i]] + er[dst[i]];
    e = (e >= 0.0f) ? e : LRELU_SLOPE * e;
    e_edge[i] = e;
    atomicMax(&emax[dst[i]], fflip(e));
}

// ---- 4: e_exp = exp(e - emax[dst]); segment sum over dst ----
__global__ void gat_edge_exp_kernel(const int* __restrict__ dst,
                                    const float* __restrict__ e_edge,
                                    const unsigned* __restrict__ emax,
                                    float* __restrict__ eexp,
                                    float* __restrict__ esum) {
    int i = blockIdx.x * blockDim.x + threadIdx.x;
    if (i >= N_EDGES) return;
    int d = dst[i];
    float ex = __expf(e_edge[i] - funflip(emax[d]));
    eexp[i] = ex;
    atomicAdd(&esum[d], ex);
}

// ---- 5: h_out[dst] += alpha * z[src]  (one thread per edge-channel) ----
__global__ void gat_edge_accum_kernel(const int* __restrict__ src,
                                      const int* __restrict__ dst,
                                      const float* __restrict__ z,
                                      const float* __restrict__ eexp,
                                      const float* __restrict__ esum,
                                      float* __restrict__ hout) {
    int gid = blockIdx.x * blockDim.x + threadIdx.x;
    int edge = gid >> 5;
    if (edge >= N_EDGES) return;
    int c = gid & 31;
    int s = src[edge], d = dst[edge];
    float alpha = eexp[edge] / esum[d];
    atomicAdd(&hout[(size_t)d * OUT_DIM + c], alpha * z[(size_t)s * OUT_DIM + c]);
}

// ---- 6: per-channel sum / sumsq for BN ----
__global__ void gat_bn_stats_kernel(const float* __restrict__ hout, float* __restrict__ bn) {
    __shared__ float ssum[OUT_DIM];
    __shared__ float ssq[OUT_DIM];
    int lane = threadIdx.x & 31;
    if (threadIdx.x < OUT_DIM) { ssum[threadIdx.x] = 0.0f; ssq[threadIdx.x] = 0.0f; }
    __syncthreads();
    int row = (blockIdx.x * blockDim.x + threadIdx.x) >> 5;
    int stride = (gridDim.x * blockDim.x) >> 5;
    float s = 0.0f, q = 0.0f;
    for (; row < N_NODES; row += stride) {
        float v = hout[(size_t)row * OUT_DIM + lane];
        s += v;
        q += v * v;
    }
    atomicAdd(&ssum[lane], s);
    atomicAdd(&ssq[lane], q);
    __syncthreads();
    if (threadIdx.x < OUT_DIM) {
        atomicAdd(&bn[threadIdx.x], ssum[threadIdx.x]);
        atomicAdd(&bn[OUT_DIM + threadIdx.x], ssq[threadIdx.x]);
    }
}

// ---- 7: BN + ELU, in place over d_out ----
__global__ void gat_bn_apply_kernel(const float* __restrict__ bn,
                                    const float* __restrict__ gamma,
                                    const float* __restrict__ beta,
                                    float* __restrict__ out) {
    int idx = blockIdx.x * blockDim.x + threadIdx.x;
    if (idx >= N_NODES * OUT_DIM) return;
    int c = idx & 31;
    const float inv_n = 1.0f / (float)N_NODES;
    float mean = bn[c] * inv_n;
    float var  = bn[OUT_DIM + c] * inv_n - mean * mean;
    float x = (out[idx] - mean) * rsqrtf(var + BN_EPS) * gamma[c] + beta[c];
    out[idx] = (x > 0.0f) ? x : expm1f(x);
}

extern "C" void kernel_launch(void* const* d_in, const int* in_sizes, int n_in,
                              void* d_out, int out_size, void* d_ws, size_t ws_size,
                              hipStream_t stream) {
    const float* h     = (const float*)d_in[0];
    const float* W_fc  = (const float*)d_in[1];
    const float* W_at  = (const float*)d_in[2];
    const float* gamma = (const float*)d_in[3];
    const float* beta  = (const float*)d_in[4];
    const int*   src   = (const int*)d_in[5];
    const int*   dst   = (const int*)d_in[6];
    float* out = (float*)d_out;

    // workspace layout
    float* z        = (float*)d_ws;                       // N*32
    float* el       = z + (size_t)N_NODES * OUT_DIM;      // N
    float* er       = el + N_NODES;                       // N
    unsigned* emax  = (unsigned*)(er + N_NODES);          // N
    float* esum     = (float*)(emax + N_NODES);           // N
    float* e_edge   = esum + N_NODES;                     // E
    float* eexp     = e_edge + N_EDGES;                   // E
    float* bn       = eexp + N_EDGES;                     // 64

    const int B = 256;

    gat_init_kernel<<<1024, B, 0, stream>>>(out, esum, emax, bn);

    // 6250 row-tiles of 16, 8 waves/block -> 782 blocks
    gat_gemm_wmma_kernel<<<(N_NODES / 16 + 7) / 8, B, 0, stream>>>(h, W_fc, z);

    gat_attn_proj_kernel<<<(N_NODES * OUT_DIM + B - 1) / B, B, 0, stream>>>(z, W_at, el, er);

    gat_edge_max_kernel<<<(N_EDGES + B - 1) / B, B, 0, stream>>>(el, er, src, dst, e_edge, emax);

    gat_edge_exp_kernel<<<(N_EDGES + B - 1) / B, B, 0, stream>>>(dst, e_edge, emax, eexp, esum);

    gat_edge_accum_kernel<<<((size_t)N_EDGES * OUT_DIM + B - 1) / B, B, 0, stream>>>(
        src, dst, z, eexp, esum, out);

    gat_bn_stats_kernel<<<512, B, 0, stream>>>(out, bn);

    gat_bn_apply_kernel<<<(N_NODES * OUT_DIM + B - 1) / B, B, 0, stream>>>(bn, gamma, beta, out);
}